// MyICP_70025146794559
// MI455X (gfx1250) — compile-verified
//
#include <hip/hip_runtime.h>
#include <math.h>

typedef __attribute__((ext_vector_type(2))) float v2f;
typedef __attribute__((ext_vector_type(8))) float v8f;

#define N_PTS    4096
#define ICP_IT   10
#define THREADS  256
#define WAVES    (THREADS / 32)
#define TILES    (N_PTS / 16)   // 256 source tiles of 16 points
#define CHUNKS   (N_PTS / 16)   // 256 target chunks of 16 points

// ---------------------------------------------------------------------------
// Thread-0 Kabsch update: red[0..2] = sum(corr), red[3..11] = sum(s_i*c_j).
// H = sum s c^T - N*sbar*cbar^T ;  R = V*diag(1,1,d)*U^T ; t = cbar - R*sbar.
// ---------------------------------------------------------------------------
__device__ __forceinline__ void kabsch_update(const float* red, const float* sMean,
                                              float* sRt) {
    const float Nf = (float)N_PTS;
    float cb[3] = {red[0] / Nf, red[1] / Nf, red[2] / Nf};
    float H[9];
#pragma unroll
    for (int i = 0; i < 3; ++i)
#pragma unroll
        for (int j = 0; j < 3; ++j)
            H[i * 3 + j] = red[3 + i * 3 + j] - Nf * sMean[i] * cb[j];

    // A = H^T H (symmetric), Jacobi eigendecomposition -> V (eigvecs), eigvals.
    float A[3][3];
#pragma unroll
    for (int i = 0; i < 3; ++i)
#pragma unroll
        for (int j = 0; j < 3; ++j)
            A[i][j] = H[0 + i] * H[0 + j] + H[3 + i] * H[3 + j] + H[6 + i] * H[6 + j];

    float V[3][3] = {{1.f, 0.f, 0.f}, {0.f, 1.f, 0.f}, {0.f, 0.f, 1.f}};
    for (int sweep = 0; sweep < 12; ++sweep) {
        for (int p = 0; p < 2; ++p) {
            for (int q = p + 1; q < 3; ++q) {
                float apq = A[p][q];
                if (fabsf(apq) < 1e-20f) continue;
                float theta = (A[q][q] - A[p][p]) / (2.f * apq);
                float tj = (theta >= 0.f ? 1.f : -1.f) /
                           (fabsf(theta) + sqrtf(1.f + theta * theta));
                float cj = 1.f / sqrtf(1.f + tj * tj);
                float sj = tj * cj;
#pragma unroll
                for (int k = 0; k < 3; ++k) {  // A <- A * J (columns p,q)
                    float akp = A[k][p], akq = A[k][q];
                    A[k][p] = cj * akp - sj * akq;
                    A[k][q] = sj * akp + cj * akq;
                }
#pragma unroll
                for (int k = 0; k < 3; ++k) {  // A <- J^T * A (rows p,q)
                    float apk = A[p][k], aqk = A[q][k];
                    A[p][k] = cj * apk - sj * aqk;
                    A[q][k] = sj * apk + cj * aqk;
                }
#pragma unroll
                for (int k = 0; k < 3; ++k) {  // V <- V * J
                    float vkp = V[k][p], vkq = V[k][q];
                    V[k][p] = cj * vkp - sj * vkq;
                    V[k][q] = sj * vkp + cj * vkq;
                }
            }
        }
    }
    float ev[3] = {A[0][0], A[1][1], A[2][2]};
    int o0 = 0, o1 = 1, o2 = 2, tmp;
    if (ev[o0] < ev[o1]) { tmp = o0; o0 = o1; o1 = tmp; }
    if (ev[o1] < ev[o2]) { tmp = o1; o1 = o2; o2 = tmp; }
    if (ev[o0] < ev[o1]) { tmp = o0; o0 = o1; o1 = tmp; }

    float v0[3] = {V[0][o0], V[1][o0], V[2][o0]};
    float v1[3] = {V[0][o1], V[1][o1], V[2][o1]};
    float v2[3] = {V[0][o2], V[1][o2], V[2][o2]};

    // U columns: u0 = norm(H v0); u1 = norm(H v1 orthogonalized); u2 = u0 x u1
    float u0[3], u1[3], u2[3];
#pragma unroll
    for (int i = 0; i < 3; ++i)
        u0[i] = H[i * 3 + 0] * v0[0] + H[i * 3 + 1] * v0[1] + H[i * 3 + 2] * v0[2];
    float n0 = sqrtf(fmaxf(u0[0] * u0[0] + u0[1] * u0[1] + u0[2] * u0[2], 1e-30f));
#pragma unroll
    for (int i = 0; i < 3; ++i) u0[i] /= n0;
#pragma unroll
    for (int i = 0; i < 3; ++i)
        u1[i] = H[i * 3 + 0] * v1[0] + H[i * 3 + 1] * v1[1] + H[i * 3 + 2] * v1[2];
    float d01 = u1[0] * u0[0] + u1[1] * u0[1] + u1[2] * u0[2];
#pragma unroll
    for (int i = 0; i < 3; ++i) u1[i] -= d01 * u0[i];
    float n1 = sqrtf(fmaxf(u1[0] * u1[0] + u1[1] * u1[1] + u1[2] * u1[2], 1e-30f));
#pragma unroll
    for (int i = 0; i < 3; ++i) u1[i] /= n1;
    u2[0] = u0[1] * u1[2] - u0[2] * u1[1];
    u2[1] = u0[2] * u1[0] - u0[0] * u1[2];
    u2[2] = u0[0] * u1[1] - u0[1] * u1[0];

    // Reflection fix folded into sign(det V)   (det U' == +1 by construction)
    float detV = v0[0] * (v1[1] * v2[2] - v1[2] * v2[1]) -
                 v1[0] * (v0[1] * v2[2] - v0[2] * v2[1]) +
                 v2[0] * (v0[1] * v1[2] - v0[2] * v1[1]);
    float d = (detV >= 0.f) ? 1.f : -1.f;

    float R[9];
#pragma unroll
    for (int i = 0; i < 3; ++i)
#pragma unroll
        for (int j = 0; j < 3; ++j)
            R[i * 3 + j] = v0[i] * u0[j] + v1[i] * u1[j] + d * v2[i] * u2[j];

#pragma unroll
    for (int k = 0; k < 9; ++k) sRt[k] = R[k];
#pragma unroll
    for (int i = 0; i < 3; ++i)
        sRt[9 + i] = cb[i] - (R[i * 3 + 0] * sMean[0] + R[i * 3 + 1] * sMean[1] +
                              R[i * 3 + 2] * sMean[2]);
}

// ---------------------------------------------------------------------------
// One workgroup per batch. NN search tiled 16x16 through V_WMMA_F32_16X16X4_F32:
// score = -2*x.y + |y|^2  (the |x|^2 row-constant cannot change the argmin).
// Two independent (even/odd chunk) argmin streams halve the select-chain and
// let the scheduler dual-issue; merged with first-index tie-break at the end.
// ---------------------------------------------------------------------------
__global__ __launch_bounds__(THREADS) void icp_wmma_kernel(
    const float* __restrict__ src, const float* __restrict__ tgt,
    float* __restrict__ out, int B) {
    __shared__ float4 tgtP[N_PTS];   // (-2tx, -2ty, -2tz, |t|^2)   64 KB
    __shared__ float4 srcP[N_PTS];   // (sx, sy, sz, 0)             64 KB
    __shared__ float red[12];
    __shared__ float sRt[12];        // R (row-major, 9) + t (3)
    __shared__ float sMean[3];

    const int b    = blockIdx.x;
    const int tid  = threadIdx.x;
    const int lane = tid & 31;
    const int wave = tid >> 5;
    const bool hi  = lane >= 16;

    const float* sb = src + (size_t)b * 3 * N_PTS;
    const float* tb = tgt + (size_t)b * 3 * N_PTS;

    if (tid == 0) {
#pragma unroll
        for (int k = 0; k < 12; ++k) red[k] = 0.f;
    }
    __syncthreads();

    // ---- pack target/source into LDS, accumulate source mean ----
    float psx = 0.f, psy = 0.f, psz = 0.f;
    for (int i = tid; i < N_PTS; i += THREADS) {
        float x = sb[i], y = sb[N_PTS + i], z = sb[2 * N_PTS + i];
        srcP[i] = make_float4(x, y, z, 0.f);
        psx += x; psy += y; psz += z;
        float tx = tb[i], ty = tb[N_PTS + i], tz = tb[2 * N_PTS + i];
        tgtP[i] = make_float4(-2.f * tx, -2.f * ty, -2.f * tz,
                              tx * tx + ty * ty + tz * tz);
    }
#pragma unroll
    for (int m = 1; m < 32; m <<= 1) {
        psx += __shfl_xor(psx, m, 32);
        psy += __shfl_xor(psy, m, 32);
        psz += __shfl_xor(psz, m, 32);
    }
    if (lane == 0) {
        atomicAdd(&red[0], psx);
        atomicAdd(&red[1], psy);
        atomicAdd(&red[2], psz);
    }
    __syncthreads();
    if (tid == 0) {
        const float invN = 1.f / (float)N_PTS;
        sMean[0] = red[0] * invN; sMean[1] = red[1] * invN; sMean[2] = red[2] * invN;
        sRt[0] = 1.f; sRt[1] = 0.f; sRt[2] = 0.f;
        sRt[3] = 0.f; sRt[4] = 1.f; sRt[5] = 0.f;
        sRt[6] = 0.f; sRt[7] = 0.f; sRt[8] = 1.f;
        sRt[9] = 0.f; sRt[10] = 0.f; sRt[11] = 0.f;
#pragma unroll
        for (int k = 0; k < 12; ++k) red[k] = 0.f;
    }
    __syncthreads();

    const float* __restrict__ tpb = (const float*)tgtP;
    const int lbase = ((lane & 15) << 2) + (hi ? 2 : 0);

    for (int iter = 0; iter < ICP_IT; ++iter) {
        const float R0 = sRt[0], R1 = sRt[1], R2 = sRt[2];
        const float R3 = sRt[3], R4 = sRt[4], R5 = sRt[5];
        const float R6 = sRt[6], R7 = sRt[7], R8 = sRt[8];
        const float T0 = sRt[9], T1 = sRt[10], T2 = sRt[11];

        float aC0 = 0.f, aC1 = 0.f, aC2 = 0.f;
        float aH[9] = {0.f, 0.f, 0.f, 0.f, 0.f, 0.f, 0.f, 0.f, 0.f};

        for (int tile = wave; tile < TILES; tile += WAVES) {
            const int row = lane & 15;
            const float4 sp = srcP[(tile << 4) + row];
            const float cx = R0 * sp.x + R1 * sp.y + R2 * sp.z + T0;
            const float cy = R3 * sp.x + R4 * sp.y + R5 * sp.z + T1;
            const float cz = R6 * sp.x + R7 * sp.y + R8 * sp.z + T2;

            // A operand 16x4 f32: lanes 0-15 -> (K0,K1)=(cx,cy);
            //                     lanes 16-31 -> (K2,K3)=(cz,1)
            v2f a;
            a[0] = hi ? cz : cx;
            a[1] = hi ? 1.0f : cy;

            float best0[8], best1[8];
            int bidx0[8], bidx1[8];
#pragma unroll
            for (int r = 0; r < 8; ++r) {
                best0[r] = 3.4e38f; bidx0[r] = 0;
                best1[r] = 3.4e38f; bidx1[r] = 0;
            }

            for (int ch = 0; ch < CHUNKS; ch += 2) {
                // B operand 4x16 f32: lane n<16 -> (K0,K1)=(-2tx,-2ty) of tgt n;
                //                     lane n+16 -> (K2,K3)=(-2tz,|t|^2)
                const float* tp0 = tpb + (ch << 6) + lbase;
                const float* tp1 = tp0 + 64;
                v2f bo0, bo1;
                bo0[0] = tp0[0]; bo0[1] = tp0[1];
                bo1[0] = tp1[0]; bo1[1] = tp1[1];
                v8f c0 = {};
                v8f d0 = __builtin_amdgcn_wmma_f32_16x16x4_f32(
                    false, a, false, bo0, (short)0, c0, false, false);
                v8f c1 = {};
                v8f d1 = __builtin_amdgcn_wmma_f32_16x16x4_f32(
                    false, a, false, bo1, (short)0, c1, false, false);
                const int t0 = (ch << 4) + (lane & 15);
                const int t1 = t0 + 16;
#pragma unroll
                for (int r = 0; r < 8; ++r) {
                    float v0 = d0[r];
                    float v1 = d1[r];
                    bool k0 = v0 < best0[r];
                    bool k1 = v1 < best1[r];
                    best0[r] = k0 ? v0 : best0[r];
                    bidx0[r] = k0 ? t0 : bidx0[r];
                    best1[r] = k1 ? v1 : best1[r];
                    bidx1[r] = k1 ? t1 : bidx1[r];
                }
            }

            // merge odd stream into even stream (smaller index wins ties)
#pragma unroll
            for (int r = 0; r < 8; ++r) {
                bool tk = (best1[r] < best0[r]) ||
                          (best1[r] == best0[r] && bidx1[r] < bidx0[r]);
                best0[r] = tk ? best1[r] : best0[r];
                bidx0[r] = tk ? bidx1[r] : bidx0[r];
            }

            // branchless argmin across the 16 lanes of each half
#pragma unroll
            for (int r = 0; r < 8; ++r) {
                float bv = best0[r];
                int bi = bidx0[r];
#pragma unroll
                for (int m = 1; m < 16; m <<= 1) {
                    float ov = __shfl_xor(bv, m, 32);
                    int oi = __shfl_xor(bi, m, 32);
                    bool tk = (ov < bv) || (ov == bv && oi < bi);
                    bv = tk ? ov : bv;
                    bi = tk ? oi : bi;
                }
                best0[r] = bv;
                bidx0[r] = bi;
            }

            // lanes 0-7 own rows 0-7 (their own src point), lanes 24-31 rows 8-15
            if (lane < 8 || lane >= 24) {
                const int r = lane & 7;
                const float4 p = tgtP[bidx0[r]];
                const float ccx = -0.5f * p.x, ccy = -0.5f * p.y, ccz = -0.5f * p.z;
                aC0 += ccx; aC1 += ccy; aC2 += ccz;
                aH[0] += sp.x * ccx; aH[1] += sp.x * ccy; aH[2] += sp.x * ccz;
                aH[3] += sp.y * ccx; aH[4] += sp.y * ccy; aH[5] += sp.y * ccz;
                aH[6] += sp.z * ccx; aH[7] += sp.z * ccy; aH[8] += sp.z * ccz;
            }
        }

        // block-wide reduction of the 12 covariance accumulators
        float vals[12] = {aC0, aC1, aC2, aH[0], aH[1], aH[2],
                          aH[3], aH[4], aH[5], aH[6], aH[7], aH[8]};
#pragma unroll
        for (int k = 0; k < 12; ++k) {
            float v = vals[k];
#pragma unroll
            for (int m = 1; m < 32; m <<= 1) v += __shfl_xor(v, m, 32);
            if (lane == 0) atomicAdd(&red[k], v);
        }
        __syncthreads();
        if (tid == 0) {
            kabsch_update(red, sMean, sRt);
#pragma unroll
            for (int k = 0; k < 12; ++k) red[k] = 0.f;
        }
        __syncthreads();
    }

    // ---- outputs: R_ab[B,3,3], t_ab[B,3], R_ba[B,3,3], t_ba[B,3] (flat) ----
    if (tid == 0) {
        float R[9], T[3];
#pragma unroll
        for (int k = 0; k < 9; ++k) R[k] = sRt[k];
#pragma unroll
        for (int k = 0; k < 3; ++k) T[k] = sRt[9 + k];

        float* oRab = out;
        float* oTab = out + B * 9;
        float* oRba = out + B * 12;
        float* oTba = out + B * 12 + B * 9;
#pragma unroll
        for (int k = 0; k < 9; ++k) oRab[b * 9 + k] = R[k];
#pragma unroll
        for (int k = 0; k < 3; ++k) oTab[b * 3 + k] = T[k];
#pragma unroll
        for (int i = 0; i < 3; ++i)
#pragma unroll
            for (int j = 0; j < 3; ++j)
                oRba[b * 9 + i * 3 + j] = R[j * 3 + i];
#pragma unroll
        for (int i = 0; i < 3; ++i)
            oTba[b * 3 + i] =
                -(R[0 * 3 + i] * T[0] + R[1 * 3 + i] * T[1] + R[2 * 3 + i] * T[2]);
    }
}

extern "C" void kernel_launch(void* const* d_in, const int* in_sizes, int n_in,
                              void* d_out, int out_size, void* d_ws, size_t ws_size,
                              hipStream_t stream) {
    const float* src = (const float*)d_in[0];
    const float* tgt = (const float*)d_in[1];
    float* out = (float*)d_out;
    const int B = in_sizes[0] / (3 * N_PTS);  // 8
    icp_wmma_kernel<<<dim3(B), dim3(THREADS), 0, stream>>>(src, tgt, out, B);
}